// GNN_11553462026250
// MI455X (gfx1250) — compile-verified
//
#include <hip/hip_runtime.h>
#include <hip/hip_bf16.h>

// ---------------------------------------------------------------------------
// GCN forward on MI455X (gfx1250, wave32).
//   layer: h = x@W ; agg = scatter_add(h[src]*dinv[src]*dinv[dst] -> dst)
//          out = relu(agg + h*dinv^2 + b)
//   pool : mean over batch segments -> [64,128] ; head: 128->64 relu -> 1
// GEMM: V_WMMA_F32_16X16X4_F32, one wave owns a 16x128 row stripe (8 C tiles)
//       so each A fragment feeds 8 WMMAs; stores are branch-free.
// Edge phase: one wave per edge, float4 per lane, L2-resident f32 atomics
// (25.6 MB feature matrix fits in the 192 MB L2).
// ---------------------------------------------------------------------------

typedef float v2f __attribute__((ext_vector_type(2)));
typedef float v8f __attribute__((ext_vector_type(8)));

#define HDIM 128
#define NGRAPH 64

// ---------------- dense GEMM: H[N,128] = X[N,128] @ W[128,128] --------------
// block = 128 (4 waves); wave w computes rows (blockIdx.x*4+w)*16 .. +15,
// all 128 columns (8 WMMA tiles). K-loop: 32 steps of K=4.
__global__ __launch_bounds__(128) void gcn_gemm_f32(const float* __restrict__ X,
                                                    const float* __restrict__ W,
                                                    float* __restrict__ H,
                                                    int N) {
  const int lane = threadIdx.x & 31;
  const int wave = threadIdx.x >> 5;
  const int m0 = (blockIdx.x * 4 + wave) * 16;

  // A-frag (16x4 f32): lanes 0-15 -> M=lane, K=k0+{0,1}; lanes 16-31 -> K=k0+{2,3}
  const int rowA  = min(m0 + (lane & 15), N - 1);   // clamp keeps EXEC all-ones
  const int khalf = (lane >> 4) * 2;                // 0 or 2
  const int colB  = lane & 15;                      // B-frag: lane -> N column

  v8f acc[8];
#pragma unroll
  for (int nt = 0; nt < 8; ++nt) acc[nt] = (v8f){};

#pragma unroll 2
  for (int k0 = 0; k0 < HDIM; k0 += 4) {
    v2f a = *(const v2f*)(X + rowA * HDIM + k0 + khalf);
    const float* wp0 = W + (k0 + khalf) * HDIM + colB;      // K   row
    const float* wp1 = wp0 + HDIM;                          // K+1 row
#pragma unroll
    for (int nt = 0; nt < 8; ++nt) {
      v2f b;
      b.x = wp0[nt * 16];
      b.y = wp1[nt * 16];
      acc[nt] = __builtin_amdgcn_wmma_f32_16x16x4_f32(false, a, false, b,
                                                      (short)0, acc[nt], false, false);
    }
  }

  // C/D layout: VGPR r -> M = r (lanes 0-15) / M = 8+r (lanes 16-31); N = lane&15
  if (m0 < N) {   // wave-uniform: tile fully in-bounds (N % 16 == 0)
    const int rC = m0 + (lane >> 4) * 8;
    const int cC = lane & 15;
#pragma unroll
    for (int nt = 0; nt < 8; ++nt)
#pragma unroll
      for (int r = 0; r < 8; ++r)
        H[(rC + r) * HDIM + nt * 16 + cC] = acc[nt][r];
  }
}

// ---------------- zero (float4) --------------------------------------------
__global__ void gcn_zero4(float4* __restrict__ p, int n4) {
  int i = blockIdx.x * blockDim.x + threadIdx.x;
  if (i < n4) p[i] = make_float4(0.f, 0.f, 0.f, 0.f);
}

// ---------------- degree / dinv --------------------------------------------
__global__ void gcn_deg(const int* __restrict__ dst, float* __restrict__ deg, int E) {
  int e = blockIdx.x * blockDim.x + threadIdx.x;
  if (e < E) atomicAdd(deg + dst[e], 1.0f);
}

__global__ void gcn_dinv(float* __restrict__ deg, int N) {
  int n = blockIdx.x * blockDim.x + threadIdx.x;
  if (n < N) deg[n] = rsqrtf(deg[n] + 1.0f);   // self-loop: deg+1
}

// ---------------- edge scatter: one wave per edge --------------------------
__global__ __launch_bounds__(256) void gcn_scatter(const float* __restrict__ H,
                                                   const int* __restrict__ src,
                                                   const int* __restrict__ dst,
                                                   const float* __restrict__ dinv,
                                                   float* __restrict__ AGG,
                                                   int E) {
  int e = blockIdx.x * 8 + (threadIdx.x >> 5);
  if (e >= E) return;
  int lane = threadIdx.x & 31;
  int s = src[e], d = dst[e];
  float norm = dinv[s] * dinv[d];
  const float4 v = *(const float4*)(H + s * HDIM + lane * 4);
  float* out = AGG + d * HDIM + lane * 4;
  atomicAdd(out + 0, v.x * norm);
  atomicAdd(out + 1, v.y * norm);
  atomicAdd(out + 2, v.z * norm);
  atomicAdd(out + 3, v.w * norm);
}

// ---------------- self-loop + bias + relu (in place on AGG, float4) --------
__global__ void gcn_combine(float4* __restrict__ AGG, const float4* __restrict__ H,
                            const float* __restrict__ dinv, const float4* __restrict__ b4,
                            int N) {
  int i = blockIdx.x * blockDim.x + threadIdx.x;   // over N * 32
  if (i >= N * (HDIM / 4)) return;
  int n = i >> 5, f = i & 31;
  float di = dinv[n];
  float di2 = di * di;
  float4 a = AGG[i], h = H[i], bb = b4[f];
  a.x = fmaxf(a.x + h.x * di2 + bb.x, 0.0f);
  a.y = fmaxf(a.y + h.y * di2 + bb.y, 0.0f);
  a.z = fmaxf(a.z + h.z * di2 + bb.z, 0.0f);
  a.w = fmaxf(a.w + h.w * di2 + bb.w, 0.0f);
  AGG[i] = a;
}

// ---------------- mean pool: wave per node ---------------------------------
__global__ __launch_bounds__(256) void gcn_pool(const float* __restrict__ H,
                                                const int* __restrict__ batch,
                                                float* __restrict__ pool,
                                                float* __restrict__ cnt, int N) {
  int n = blockIdx.x * 8 + (threadIdx.x >> 5);
  if (n >= N) return;
  int lane = threadIdx.x & 31;
  int g = batch[n];
  const float4 v = *(const float4*)(H + n * HDIM + lane * 4);
  float* p = pool + g * HDIM + lane * 4;
  atomicAdd(p + 0, v.x);
  atomicAdd(p + 1, v.y);
  atomicAdd(p + 2, v.z);
  atomicAdd(p + 3, v.w);
  if (lane == 0) atomicAdd(cnt + g, 1.0f);
}

// ---------------- MLP head: one block per graph ----------------------------
__global__ __launch_bounds__(64) void gcn_head(const float* __restrict__ pool,
                                               const float* __restrict__ cnt,
                                               const float* __restrict__ Wfc1,
                                               const float* __restrict__ bfc1,
                                               const float* __restrict__ Wfc2,
                                               const float* __restrict__ bfc2,
                                               float* __restrict__ out) {
  __shared__ float mean[HDIM];
  __shared__ float s1[64];
  int g = blockIdx.x, t = threadIdx.x;
  float inv = 1.0f / fmaxf(cnt[g], 1.0f);
  mean[t]      = pool[g * HDIM + t] * inv;
  mean[t + 64] = pool[g * HDIM + 64 + t] * inv;
  __syncthreads();
  float s = bfc1[t];
#pragma unroll 8
  for (int k = 0; k < HDIM; ++k) s += mean[k] * Wfc1[k * 64 + t];
  s1[t] = fmaxf(s, 0.0f);
  __syncthreads();
  if (t == 0) {
    float o = bfc2[0];
#pragma unroll 8
    for (int j = 0; j < 64; ++j) o += s1[j] * Wfc2[j];
    out[g] = o;
  }
}

// ---------------------------------------------------------------------------
extern "C" void kernel_launch(void* const* d_in, const int* in_sizes, int n_in,
                              void* d_out, int out_size, void* d_ws, size_t ws_size,
                              hipStream_t stream) {
  const float* x     = (const float*)d_in[0];
  const int*   eidx  = (const int*)d_in[1];     // [2, E] flat
  const int*   batch = (const int*)d_in[2];
  const float* W1    = (const float*)d_in[3];
  const float* b1    = (const float*)d_in[4];
  const float* W2    = (const float*)d_in[5];
  const float* b2    = (const float*)d_in[6];
  const float* Wfc1  = (const float*)d_in[7];
  const float* bfc1  = (const float*)d_in[8];
  const float* Wfc2  = (const float*)d_in[9];
  const float* bfc2  = (const float*)d_in[10];
  float* out = (float*)d_out;

  const int N = in_sizes[0] / HDIM;   // 50000
  const int E = in_sizes[1] / 2;      // 600000
  const int* src = eidx;
  const int* dst = eidx + E;

  // workspace layout (floats)
  float* ws    = (float*)d_ws;
  float* dinv  = ws;                                  // N
  float* bufA  = ws + ((N + 15) & ~15);               // N*128
  float* bufB  = bufA + (size_t)N * HDIM;             // N*128
  float* pool  = bufB + (size_t)N * HDIM;             // 64*128
  float* cnt   = pool + NGRAPH * HDIM;                // 64

  const int mt = (N + 15) / 16;                       // 3125
  const dim3 gemmGrid((mt + 3) / 4), gemmBlk(128);    // 4 row-stripe waves/block
  const int feB = 256;
  const int nh4 = N * (HDIM / 4);
  const int feGrid4 = (nh4 + feB - 1) / feB;
  const int edgeWaveGrid = (E + 7) / 8;
  const int nodeWaveGrid = (N + 7) / 8;

  // degree -> dinv
  gcn_zero4<<<(N / 4 + 255) / 256, 256, 0, stream>>>((float4*)dinv, N / 4);
  gcn_deg<<<(E + 255) / 256, 256, 0, stream>>>(dst, dinv, E);
  gcn_dinv<<<(N + 255) / 256, 256, 0, stream>>>(dinv, N);

  // ---- layer 1 ----
  gcn_gemm_f32<<<gemmGrid, gemmBlk, 0, stream>>>(x, W1, bufA, N);
  gcn_zero4<<<feGrid4, feB, 0, stream>>>((float4*)bufB, nh4);
  gcn_scatter<<<edgeWaveGrid, 256, 0, stream>>>(bufA, src, dst, dinv, bufB, E);
  gcn_combine<<<feGrid4, feB, 0, stream>>>((float4*)bufB, (const float4*)bufA,
                                           dinv, (const float4*)b1, N);

  // ---- layer 2 ----
  gcn_gemm_f32<<<gemmGrid, gemmBlk, 0, stream>>>(bufB, W2, bufA, N);
  gcn_zero4<<<feGrid4, feB, 0, stream>>>((float4*)bufB, nh4);
  gcn_scatter<<<edgeWaveGrid, 256, 0, stream>>>(bufA, src, dst, dinv, bufB, E);
  gcn_combine<<<feGrid4, feB, 0, stream>>>((float4*)bufB, (const float4*)bufA,
                                           dinv, (const float4*)b2, N);

  // ---- pool + head ----
  gcn_zero4<<<((NGRAPH * HDIM + NGRAPH) / 4 + 255) / 256, 256, 0, stream>>>(
      (float4*)pool, (NGRAPH * HDIM + NGRAPH) / 4);
  gcn_pool<<<nodeWaveGrid, 256, 0, stream>>>(bufB, batch, pool, cnt, N);
  gcn_head<<<NGRAPH, 64, 0, stream>>>(pool, cnt, Wfc1, bfc1, Wfc2, bfc2, out);
}